// BinaryTreeConv_69157563400280
// MI455X (gfx1250) — compile-verified
//
#include <hip/hip_runtime.h>

typedef float v2f __attribute__((ext_vector_type(2)));
typedef float v8f __attribute__((ext_vector_type(8)));

#define B_      64
#define CIN     256
#define COUT    256
#define NNODES  2048
#define KDIM    (CIN * 3)   // 768
#define NT      16          // output columns per block
#define THREADS 256         // 8 waves (wave32)

// One block: batch b, 16 output columns, all 256 output rows.
// Stage gathered panel G[KDIM][NT] in LDS (48 KB), shared by 8 waves.
// Wave w computes M-tiles [32w,32w+16) and [32w+16,32w+32) via
// v_wmma_f32_16x16x4_f32, K-loop of 192 steps.
__global__ __launch_bounds__(THREADS)
void btc_wmma_kernel(const float* __restrict__ trees,
                     const void*  __restrict__ idx_raw,
                     int idx_is64,
                     const float* __restrict__ W,
                     float* __restrict__ out)
{
    __shared__ float panel[KDIM * NT]; // 48 KB, layout [k][col]

    const int b  = blockIdx.y;
    const int n0 = blockIdx.x * NT;
    const int t  = threadIdx.x;

    const float* treesB = trees + (size_t)b * CIN * NNODES;
    const int*       idx32 = (const int*)idx_raw;
    const long long* idx64 = (const long long*)idx_raw;

    // ---- Gather fill: panel[kk*16+col] = trees_pad[b, c, idx[b, 3*(n0+col)+j]]
    // where kk = c*3 + j (matches W row flattening). Linear LDS stores
    // (thread t -> word e), conflict-free.
    for (int e = t; e < KDIM * NT; e += THREADS) {
        int kk  = e >> 4;
        int col = e & 15;
        int c   = kk / 3;
        int j   = kk - 3 * c;
        long long flat = (long long)b * (3 * NNODES) + 3 * (n0 + col) + j;
        long long node = idx_is64 ? idx64[flat] : (long long)idx32[flat];
        float v = 0.0f;
        if (node > 0) v = treesB[(size_t)c * NNODES + (size_t)(node - 1)];
        panel[e] = v;
    }
    __syncthreads();

    // ---- WMMA compute
    const int wave = t >> 5;       // 0..7
    const int lane = t & 31;
    const int lh   = lane >> 4;    // lane-half: 0 or 1
    const int lm   = lane & 15;    // M (for A) / N (for B,D) position

    const int o0 = wave * 32 + lm;        // rows of M-tile 0
    const int o1 = wave * 32 + 16 + lm;   // rows of M-tile 1
    const float* W0 = W + (size_t)o0 * KDIM;
    const float* W1 = W + (size_t)o1 * KDIM;

    v8f acc0 = {0.f, 0.f, 0.f, 0.f, 0.f, 0.f, 0.f, 0.f};
    v8f acc1 = {0.f, 0.f, 0.f, 0.f, 0.f, 0.f, 0.f, 0.f};

    const int koff = 2 * lh; // per-lane-half K skew (ISA 16x16x4 layout)

#pragma unroll 8
    for (int kb = 0; kb < KDIM; kb += 4) {
        const int ka = kb + koff;

        // A 16x4: VGPR v, lane-half h -> K = kb + v + 2h, M = lane%16
        v2f a0, a1;
        a0.x = W0[ka];     a0.y = W0[ka + 1];
        a1.x = W1[ka];     a1.y = W1[ka + 1];

        // B 4x16 (mirror of A): VGPR v -> K = kb + v + 2h, N = lane%16
        // LDS banks: lanes0-15 hit {s..s+15}, lanes16-31 {s+32..s+47} -> no conflict
        v2f bb;
        bb.x = panel[(ka + 0) * NT + lm];
        bb.y = panel[(ka + 1) * NT + lm];

        acc0 = __builtin_amdgcn_wmma_f32_16x16x4_f32(
                   false, a0, false, bb, (short)0, acc0, false, false);
        acc1 = __builtin_amdgcn_wmma_f32_16x16x4_f32(
                   false, a1, false, bb, (short)0, acc1, false, false);
    }

    // ---- Store D: VGPR r -> M = r + 8*lh, N = lm
    float* outB = out + (size_t)b * COUT * NNODES + n0 + lm;
#pragma unroll
    for (int r = 0; r < 8; ++r) {
        int m0 = wave * 32 + r + 8 * lh;        // tile 0 row
        int m1 = wave * 32 + 16 + r + 8 * lh;   // tile 1 row
        outB[(size_t)m0 * NNODES] = acc0[r];
        outB[(size_t)m1 * NNODES] = acc1[r];
    }
}

extern "C" void kernel_launch(void* const* d_in, const int* in_sizes, int n_in,
                              void* d_out, int out_size, void* d_ws, size_t ws_size,
                              hipStream_t stream)
{
    const float* trees    = (const float*)d_in[0];
    const void*  idxes    = d_in[1];
    const void*  zero_pos = d_in[2];
    const float* W        = (const float*)d_in[3];
    float*       out      = (float*)d_out;

    const size_t RES_ELEMS = (size_t)B_ * COUT * NNODES; // 33,554,432 floats
    const size_t N_IDX     = (size_t)in_sizes[1];        // 393,216
    const size_t N_ZP      = (size_t)in_sizes[2];        // 64

    // int32 indices -> passthrough ints occupy 1 float slot each;
    // int64 (JAX x64 enabled) -> 2 slots each. Detect from out_size.
    const size_t sz32 = RES_ELEMS + N_IDX + N_ZP;
    const int idx_is64 = ((size_t)out_size > sz32) ? 1 : 0;
    const size_t elt = idx_is64 ? 8 : 4;

    dim3 grid(NNODES / NT, B_);   // 128 x 64
    dim3 block(THREADS);
    btc_wmma_kernel<<<grid, block, 0, stream>>>(trees, idxes, idx_is64, W, out);

    // Tuple passthrough outputs, concatenated after the GEMM result.
    char* dst = (char*)d_out + RES_ELEMS * sizeof(float);
    hipMemcpyAsync(dst, idxes, N_IDX * elt, hipMemcpyDeviceToDevice, stream);
    hipMemcpyAsync(dst + N_IDX * elt, zero_pos, N_ZP * elt,
                   hipMemcpyDeviceToDevice, stream);
}